// MultiFlashHypothesis_13993003450374
// MI455X (gfx1250) — compile-verified
//
#include <hip/hip_runtime.h>

// ---------------------------------------------------------------------------
// MultiFlashHypothesis on MI455X (gfx1250):
//   pred[c, p] = sum_i q_i * vis[vox(point_i shifted by clip(dx_c)), p]
// Reformulated as dense GEMM  pred = H(128 x 262144) * vis(262144 x 180)
// running on V_WMMA_F32_16X16X4_F32 (f32 matrix path == reference precision).
// Phases:
//   0) zero histogram workspace
//   1) scatter: atomicAdd q into per-cluster voxel histogram H
//   2) WMMA GEMM, K split 64 ways; vis/H double-buffered through LDS with
//      register prefetch so global latency overlaps the WMMA loop
//   3) reduce K-split partials into d_out
// ---------------------------------------------------------------------------

#define NXv 64
#define NYv 64
#define NZv 64
#define N_VOX (NXv * NYv * NZv)      // 262144
#define N_PMT 180
#define N_PMT_PAD 192
#define N_CLUSTERS 128
#define TOTAL_PTS (N_CLUSTERS * 8192)

#define KSPLIT 64
#define KCHUNK (N_VOX / KSPLIT)      // 4096 voxels per workgroup
#define KPANEL 64                    // voxels staged in LDS per stage
#define NPANELS (KCHUNK / KPANEL)    // 64
#define BSTRIDE 200                  // LDS row stride for B (bank-conflict free)
#define ASTRIDE 68                   // LDS row stride for A (bank-conflict free)
#define B_ELEMS (KPANEL * N_PMT_PAD) // 12288 -> 32 per thread at 384 threads
#define A_ELEMS (16 * KPANEL)        // 1024  -> <=3 per thread at 384 threads

typedef float v2f __attribute__((ext_vector_type(2)));
typedef float v8f __attribute__((ext_vector_type(8)));

__device__ __forceinline__ int clampi(int v, int lo, int hi) {
    return v < lo ? lo : (v > hi ? hi : v);
}

// ---------------------------------------------------------------- phase 0 ---
__global__ void zero_f32(float* __restrict__ p, long n) {
    long i = (long)blockIdx.x * blockDim.x + threadIdx.x;
    long stride = (long)gridDim.x * blockDim.x;
    for (; i < n; i += stride) p[i] = 0.0f;
}

// ---------------------------------------------------------------- phase 1 ---
__global__ void scatter_hist(const float* __restrict__ dx,
                             const float* __restrict__ batch,
                             const int* __restrict__ sizes,
                             const float* __restrict__ ranges,
                             float* __restrict__ h) {
    __shared__ int cum[N_CLUSTERS + 1];
    if (threadIdx.x == 0) {
        int s = 0;
        for (int c = 0; c < N_CLUSTERS; ++c) { cum[c] = s; s += sizes[c]; }
        cum[N_CLUSTERS] = s;
    }
    __syncthreads();

    int i = blockIdx.x * blockDim.x + threadIdx.x;
    if (i >= TOTAL_PTS) return;

    // binary search for segment id (handles non-uniform sizes)
    int lo = 0, hi = N_CLUSTERS;
    while (hi - lo > 1) {
        int mid = (lo + hi) >> 1;
        if (cum[mid] <= i) lo = mid; else hi = mid;
    }
    const int c = lo;

    float d = dx[c];
    d = fminf(fmaxf(d, ranges[2 * c]), ranges[2 * c + 1]);

    float4 pt = ((const float4*)batch)[i];
    int ix = clampi((int)floorf(pt.x + d), 0, NXv - 1);
    int iy = clampi((int)floorf(pt.y),     0, NYv - 1);
    int iz = clampi((int)floorf(pt.z),     0, NZv - 1);
    int vox = ix + NXv * (iy + NYv * iz);

    atomicAdd(&h[(size_t)c * N_VOX + vox], pt.w);
}

// ---------------------------------------------------------------- phase 2 ---
// grid = (KSPLIT, N_CLUSTERS/16); block = 384 (12 waves, one wave per N-tile)
__global__ __launch_bounds__(384)
void hist_gemm(const float* __restrict__ h,
               const float* __restrict__ vis,
               float* __restrict__ partial) {
    // double-buffered panels: 2*(50 KB + 4.25 KB) = 108.5 KB of 320 KB WGP LDS
    __shared__ float lds_b[2][KPANEL * BSTRIDE];
    __shared__ float lds_a[2][16 * ASTRIDE];

    const int split = blockIdx.x;
    const int c0    = blockIdx.y * 16;
    const int tid   = threadIdx.x;
    const int w     = tid >> 5;        // N-tile 0..11 (wave id)
    const int lane  = tid & 31;
    const int lm    = lane & 15;
    const int hb    = lane >> 4;       // half-wave: 0 or 1
    const int kbase = split * KCHUNK;

    v8f acc;
#pragma unroll
    for (int i = 0; i < 8; ++i) acc[i] = 0.0f;

    float rb[B_ELEMS / 384];           // 32 regs: next B panel
    float ra[3];                       // next A panel (last slot partial)

    // register-prefetch a panel's worth of global data
    auto fetch_panel = [&](int k0) {
#pragma unroll
        for (int j = 0; j < B_ELEMS / 384; ++j) {
            int idx = tid + j * 384;
            int r = idx / N_PMT_PAD;
            int n = idx - r * N_PMT_PAD;
            rb[j] = (n < N_PMT) ? vis[(size_t)(k0 + r) * N_PMT + n] : 0.0f;
        }
#pragma unroll
        for (int j = 0; j < 3; ++j) {
            int idx = tid + j * 384;
            if (idx < A_ELEMS) {
                int m = idx >> 6;
                int k = idx & 63;
                ra[j] = h[(size_t)(c0 + m) * N_VOX + k0 + k];
            }
        }
    };
    // commit prefetched registers into LDS buffer `b`
    auto commit_panel = [&](int b) {
#pragma unroll
        for (int j = 0; j < B_ELEMS / 384; ++j) {
            int idx = tid + j * 384;
            int r = idx / N_PMT_PAD;
            int n = idx - r * N_PMT_PAD;
            lds_b[b][r * BSTRIDE + n] = rb[j];
        }
#pragma unroll
        for (int j = 0; j < 3; ++j) {
            int idx = tid + j * 384;
            if (idx < A_ELEMS) {
                int m = idx >> 6;
                int k = idx & 63;
                lds_a[b][m * ASTRIDE + k] = ra[j];
            }
        }
    };

    int buf = 0;
    fetch_panel(kbase);
    commit_panel(buf);
    __syncthreads();

    for (int panel = 0; panel < NPANELS; ++panel) {
        if (panel + 1 < NPANELS)
            fetch_panel(kbase + (panel + 1) * KPANEL);   // loads fly during WMMAs

        const float* Ab = &lds_a[buf][lm * ASTRIDE + 2 * hb];
        const float* Bb = &lds_b[buf][(2 * hb) * BSTRIDE + w * 16 + lm];

#pragma unroll 4
        for (int kk = 0; kk < KPANEL; kk += 4) {
            // A 16x4 f32: lane L -> M=L%16, VGPR j holds K = 2*(L/16)+j
            v2f a;
            a.x = Ab[kk];
            a.y = Ab[kk + 1];
            // B 4x16 f32: lane L -> N=L%16, VGPR j holds K = 2*(L/16)+j
            v2f b;
            b.x = Bb[kk * BSTRIDE];
            b.y = Bb[(kk + 1) * BSTRIDE];
            // D = A*B + C  (neg_a, A, neg_b, B, c_mod, C, reuse_a, reuse_b)
            acc = __builtin_amdgcn_wmma_f32_16x16x4_f32(
                false, a, false, b, (short)0, acc, false, false);
        }

        if (panel + 1 < NPANELS) {
            __syncthreads();           // all waves done reading lds[buf^1] & [buf]
            commit_panel(buf ^ 1);
            __syncthreads();           // new panel visible
            buf ^= 1;
        }
    }

    // C/D 16x16 f32: VGPR i, lane L -> M = i + 8*(L/16), N = L%16
#pragma unroll
    for (int i = 0; i < 8; ++i) {
        int M = i + 8 * hb;
        partial[((size_t)split * N_CLUSTERS + c0 + M) * N_PMT_PAD + w * 16 + lm] = acc[i];
    }
}

// ---------------------------------------------------------------- phase 3 ---
__global__ void reduce_partials(const float* __restrict__ partial,
                                float* __restrict__ out) {
    int i = blockIdx.x * blockDim.x + threadIdx.x;   // over 128*180
    if (i >= N_CLUSTERS * N_PMT) return;
    int c = i / N_PMT;
    int p = i - c * N_PMT;
    float s = 0.0f;
#pragma unroll 4
    for (int sp = 0; sp < KSPLIT; ++sp)
        s += partial[((size_t)sp * N_CLUSTERS + c) * N_PMT_PAD + p];
    out[i] = s;
}

// ------------------------------------------------- fallback (small d_ws) ----
// Direct gather-reduce: block=192 (lane owns one PMT), grid=(clusters, splits)
__global__ __launch_bounds__(192)
void direct_fallback(const float* __restrict__ dx,
                     const float* __restrict__ batch,
                     const int* __restrict__ sizes,
                     const float* __restrict__ ranges,
                     const float* __restrict__ vis,
                     float* __restrict__ out) {
    __shared__ int cum[N_CLUSTERS + 1];
    if (threadIdx.x == 0) {
        int s = 0;
        for (int c = 0; c < N_CLUSTERS; ++c) { cum[c] = s; s += sizes[c]; }
        cum[N_CLUSTERS] = s;
    }
    __syncthreads();

    const int c   = blockIdx.x;
    const int sp  = blockIdx.y;
    const int nsp = gridDim.y;
    const int p0 = cum[c], p1 = cum[c + 1];
    const int per = (p1 - p0 + nsp - 1) / nsp;
    const int s0 = p0 + sp * per;
    const int s1 = (s0 + per < p1) ? (s0 + per) : p1;

    const float d = fminf(fmaxf(dx[c], ranges[2 * c]), ranges[2 * c + 1]);
    const int p = threadIdx.x;

    float acc = 0.0f;
    for (int i = s0; i < s1; ++i) {
        float4 pt = ((const float4*)batch)[i];
        int ix = clampi((int)floorf(pt.x + d), 0, NXv - 1);
        int iy = clampi((int)floorf(pt.y),     0, NYv - 1);
        int iz = clampi((int)floorf(pt.z),     0, NZv - 1);
        int vox = ix + NXv * (iy + NYv * iz);
        if (p < N_PMT) acc += pt.w * vis[(size_t)vox * N_PMT + p];
    }
    if (p < N_PMT) atomicAdd(&out[c * N_PMT + p], acc);
}

// ---------------------------------------------------------------------------
extern "C" void kernel_launch(void* const* d_in, const int* in_sizes, int n_in,
                              void* d_out, int out_size, void* d_ws, size_t ws_size,
                              hipStream_t stream) {
    (void)in_sizes; (void)n_in; (void)out_size;

    const float* dx     = (const float*)d_in[0];
    const float* batch  = (const float*)d_in[1];
    const int*   sizes  = (const int*)d_in[2];
    const float* ranges = (const float*)d_in[3];
    const float* vis    = (const float*)d_in[4];
    float*       out    = (float*)d_out;

    const size_t hist_bytes = (size_t)N_CLUSTERS * N_VOX * sizeof(float);              // 134.2 MB
    const size_t part_bytes = (size_t)KSPLIT * N_CLUSTERS * N_PMT_PAD * sizeof(float); // 6.3 MB

    if (ws_size >= hist_bytes + part_bytes) {
        float* h = (float*)d_ws;
        float* P = (float*)((char*)d_ws + hist_bytes);

        zero_f32<<<4096, 256, 0, stream>>>(h, (long)N_CLUSTERS * N_VOX);
        scatter_hist<<<TOTAL_PTS / 256, 256, 0, stream>>>(dx, batch, sizes, ranges, h);
        dim3 g(KSPLIT, N_CLUSTERS / 16);
        hist_gemm<<<g, 384, 0, stream>>>(h, vis, P);
        reduce_partials<<<(N_CLUSTERS * N_PMT + 255) / 256, 256, 0, stream>>>(P, out);
    } else {
        zero_f32<<<(N_CLUSTERS * N_PMT + 255) / 256, 256, 0, stream>>>(
            out, (long)N_CLUSTERS * N_PMT);
        dim3 g(N_CLUSTERS, 8);
        direct_fallback<<<g, 192, 0, stream>>>(dx, batch, sizes, ranges, vis, out);
    }
}